// BasisCustLinear_8014408974785
// MI455X (gfx1250) — compile-verified
//
#include <hip/hip_runtime.h>
#include <hip/hip_bf16.h>

// ---------------------------------------------------------------------------
// BasisCustLinear for MI455X (gfx1250, wave32)
//   out[b,o] = sum_{k=0..3} coef4[b,k] * (x @ W4_k)[b,o]
//   W4_0..2 = W3 bases repacked [IN,OUT], W4_3 = b3 reshaped, coef4[...,3] = 1
// GEMM operands are pre-swizzled into V_WMMA_F32_16X16X4_F32 fragment order
// so the inner loop is pure b128 loads + WMMAs.
// ---------------------------------------------------------------------------

#define BATCH    1024
#define IN_DIM   512
#define OUT_DIM  512
#define EMB_DIM  64
#define KQ       64
#define NBASES   3
#define MAT_ELEMS (IN_DIM * OUT_DIM)   // 262144 = 2^18

typedef float v2f __attribute__((ext_vector_type(2)));
typedef float v4f __attribute__((ext_vector_type(4)));
typedef float v8f __attribute__((ext_vector_type(8)));

// ---------------------------------------------------------------------------
// Kernel 1: per-sample coefficient MLP (tiny; thread-per-sample)
// ---------------------------------------------------------------------------
__global__ __launch_bounds__(256) void coef_kernel(
    const int* __restrict__ idxA, const int* __restrict__ idxC,
    const float* __restrict__ embA, const float* __restrict__ embC,
    const float* __restrict__ W1, const float* __restrict__ b1,
    const float* __restrict__ W2,
    float* __restrict__ coef4)
{
    int b = blockIdx.x * blockDim.x + threadIdx.x;
    if (b >= BATCH) return;

    const float* ea = embA + idxA[b] * EMB_DIM;
    const float* ec = embC + idxC[b] * EMB_DIM;

    float l0 = 0.f, l1 = 0.f, l2 = 0.f;
    for (int q = 0; q < KQ; ++q) {
        float acc = b1[q];
        const float* w = W1 + q * (2 * EMB_DIM);
#pragma unroll 8
        for (int e = 0; e < EMB_DIM; ++e)
            acc += w[e] * ea[e] + w[EMB_DIM + e] * ec[e];
        float h = tanhf(acc);
        l0 += W2[0 * KQ + q] * h;
        l1 += W2[1 * KQ + q] * h;
        l2 += W2[2 * KQ + q] * h;
    }
    float mx = fmaxf(l0, fmaxf(l1, l2));
    float e0 = __expf(l0 - mx), e1 = __expf(l1 - mx), e2 = __expf(l2 - mx);
    float inv = 1.0f / (e0 + e1 + e2);
    coef4[b * 4 + 0] = e0 * inv;
    coef4[b * 4 + 1] = e1 * inv;
    coef4[b * 4 + 2] = e2 * inv;
    coef4[b * 4 + 3] = 1.0f;
}

// ---------------------------------------------------------------------------
// Fragment swizzle helpers (ISA 7.12.2, 32-bit WMMA, wave32):
//   A 16x4 : lane = m + 16g ; VGPR0/1 hold K = 2g, 2g+1 of row m
//   B 4x16 : VGPR v, group g -> K = v + 2g, N = lane&15
// We pack TWO consecutive K-chunks (8 K values) per b128:
//   t=0,1 -> chunk even (k=8*c2)   : K = 8*c2     + 2g + t
//   t=2,3 -> chunk odd  (k=8*c2+4) : K = 8*c2 + 4 + 2g + (t-2)
// ---------------------------------------------------------------------------
__device__ __forceinline__ int frag_K(int c2, int g, int t) {
    return 8 * c2 + 4 * (t >> 1) + 2 * g + (t & 1);
}

// Kernel 2a: swizzle W3 (stride-3) + b3 into Bs:
//   Bs[j*2^18 + c2*4096 + g*2048 + n*4 + t] = W4_j[K(c2,g,t)][n]
__global__ __launch_bounds__(256) void repack_B_kernel(
    const float* __restrict__ W3, const float* __restrict__ b3,
    float* __restrict__ Bs)
{
    int idx = blockIdx.x * blockDim.x + threadIdx.x;   // 0 .. 4*2^18
    if (idx >= 4 * MAT_ELEMS) return;
    int t  = idx & 3;
    int n  = (idx >> 2) & (OUT_DIM - 1);
    int g  = (idx >> 11) & 1;
    int c2 = (idx >> 12) & 63;
    int j  = idx >> 18;
    int K  = frag_K(c2, g, t);
    int src = K * OUT_DIM + n;
    Bs[idx] = (j < NBASES) ? W3[src * NBASES + j] : b3[src];
}

// Kernel 2b: swizzle x into Xs:
//   Xs[row*512 + c2*8 + g*4 + t] = x[row][K(c2,g,t)]
__global__ __launch_bounds__(256) void repack_A_kernel(
    const float* __restrict__ x, float* __restrict__ Xs)
{
    int idx = blockIdx.x * blockDim.x + threadIdx.x;   // 0 .. 2^19
    if (idx >= BATCH * IN_DIM) return;
    int t   = idx & 3;
    int g   = (idx >> 2) & 1;
    int c2  = (idx >> 3) & 63;
    int row = idx >> 9;
    Xs[idx] = x[row * IN_DIM + frag_K(c2, g, t)];
}

// ---------------------------------------------------------------------------
// Kernel 3: fused 4-way GEMM. One wave owns a 32x16 output tile
// (2 M-subtiles -> halves basis-matrix L2 traffic); 8 f32 accumulators.
// Inner 8-K step: 2 A b128 + 4 B b128 loads, 16 V_WMMA_F32_16X16X4_F32.
// ---------------------------------------------------------------------------
__global__ __launch_bounds__(128) void gemm_basis(
    const float* __restrict__ Xs,
    const float* __restrict__ Bs,
    const float* __restrict__ coef4,
    float* __restrict__ out)
{
    const int lane   = threadIdx.x & 31;
    const int waveId = blockIdx.x * 4 + (threadIdx.x >> 5);
    const int mBlock = waveId >> 5;   // 32 M-blocks of 32 rows
    const int tileN  = waveId & 31;   // 32 col tiles of 16

    const int m = lane & 15;
    const int g = lane >> 4;
    const int row0 = mBlock * 32 + m;        // M subtile 0
    const int row1 = row0 + 16;              // M subtile 1
    const int colbase = tileN * 16;

    v8f acc00 = {}, acc01 = {}, acc02 = {}, acc03 = {};  // subtile 0, bases 0..3
    v8f acc10 = {}, acc11 = {}, acc12 = {}, acc13 = {};  // subtile 1, bases 0..3

    const float* pa0 = Xs + row0 * IN_DIM + g * 4;
    const float* pa1 = Xs + row1 * IN_DIM + g * 4;
    const float* pb0 = Bs + 0 * MAT_ELEMS + g * 2048 + (colbase + m) * 4;
    const float* pb1 = Bs + 1 * MAT_ELEMS + g * 2048 + (colbase + m) * 4;
    const float* pb2 = Bs + 2 * MAT_ELEMS + g * 2048 + (colbase + m) * 4;
    const float* pb3 = Bs + 3 * MAT_ELEMS + g * 2048 + (colbase + m) * 4;

#pragma unroll 2
    for (int c2 = 0; c2 < IN_DIM / 8; ++c2) {          // 64 iterations
        v4f a0 = *(const v4f*)(pa0 + c2 * 8);
        v4f a1 = *(const v4f*)(pa1 + c2 * 8);
        v4f b0 = *(const v4f*)(pb0 + c2 * 4096);
        v4f b1 = *(const v4f*)(pb1 + c2 * 4096);
        v4f b2 = *(const v4f*)(pb2 + c2 * 4096);
        v4f b3 = *(const v4f*)(pb3 + c2 * 4096);

        v2f a0lo = {a0.x, a0.y}, a0hi = {a0.z, a0.w};
        v2f a1lo = {a1.x, a1.y}, a1hi = {a1.z, a1.w};
        v2f b0lo = {b0.x, b0.y}, b0hi = {b0.z, b0.w};
        v2f b1lo = {b1.x, b1.y}, b1hi = {b1.z, b1.w};
        v2f b2lo = {b2.x, b2.y}, b2hi = {b2.z, b2.w};
        v2f b3lo = {b3.x, b3.y}, b3hi = {b3.z, b3.w};

        acc00 = __builtin_amdgcn_wmma_f32_16x16x4_f32(false, a0lo, false, b0lo, (short)0, acc00, false, false);
        acc01 = __builtin_amdgcn_wmma_f32_16x16x4_f32(false, a0lo, false, b1lo, (short)0, acc01, false, false);
        acc02 = __builtin_amdgcn_wmma_f32_16x16x4_f32(false, a0lo, false, b2lo, (short)0, acc02, false, false);
        acc03 = __builtin_amdgcn_wmma_f32_16x16x4_f32(false, a0lo, false, b3lo, (short)0, acc03, false, false);
        acc10 = __builtin_amdgcn_wmma_f32_16x16x4_f32(false, a1lo, false, b0lo, (short)0, acc10, false, false);
        acc11 = __builtin_amdgcn_wmma_f32_16x16x4_f32(false, a1lo, false, b1lo, (short)0, acc11, false, false);
        acc12 = __builtin_amdgcn_wmma_f32_16x16x4_f32(false, a1lo, false, b2lo, (short)0, acc12, false, false);
        acc13 = __builtin_amdgcn_wmma_f32_16x16x4_f32(false, a1lo, false, b3lo, (short)0, acc13, false, false);

        acc00 = __builtin_amdgcn_wmma_f32_16x16x4_f32(false, a0hi, false, b0hi, (short)0, acc00, false, false);
        acc01 = __builtin_amdgcn_wmma_f32_16x16x4_f32(false, a0hi, false, b1hi, (short)0, acc01, false, false);
        acc02 = __builtin_amdgcn_wmma_f32_16x16x4_f32(false, a0hi, false, b2hi, (short)0, acc02, false, false);
        acc03 = __builtin_amdgcn_wmma_f32_16x16x4_f32(false, a0hi, false, b3hi, (short)0, acc03, false, false);
        acc10 = __builtin_amdgcn_wmma_f32_16x16x4_f32(false, a1hi, false, b0hi, (short)0, acc10, false, false);
        acc11 = __builtin_amdgcn_wmma_f32_16x16x4_f32(false, a1hi, false, b1hi, (short)0, acc11, false, false);
        acc12 = __builtin_amdgcn_wmma_f32_16x16x4_f32(false, a1hi, false, b2hi, (short)0, acc12, false, false);
        acc13 = __builtin_amdgcn_wmma_f32_16x16x4_f32(false, a1hi, false, b3hi, (short)0, acc13, false, false);
    }

    // Epilogue: per accumulator row r -> M = r + 8g within each subtile.
#pragma unroll
    for (int r = 0; r < 8; ++r) {
        int rowA = mBlock * 32 + r + 8 * g;
        int rowB = rowA + 16;
        float4 cA = *(const float4*)(coef4 + rowA * 4);
        float4 cB = *(const float4*)(coef4 + rowB * 4);
        float vA = acc00[r] * cA.x + acc01[r] * cA.y + acc02[r] * cA.z + acc03[r] * cA.w;
        float vB = acc10[r] * cB.x + acc11[r] * cB.y + acc12[r] * cB.z + acc13[r] * cB.w;
        out[rowA * OUT_DIM + colbase + m] = vA;
        out[rowB * OUT_DIM + colbase + m] = vB;
    }
}

// ---------------------------------------------------------------------------
// Launch. Inputs (setup_inputs order):
//  0:x 1:idx_author 2:idx_citation 3:emb_author 4:emb_citation
//  5:W1 6:b1 7:W2 8:W3 9:b3
// Workspace: [0,4MB) Bs | [4MB,6MB) Xs | [6MB,+16KB) coef4   (~6.02 MB)
// ---------------------------------------------------------------------------
extern "C" void kernel_launch(void* const* d_in, const int* in_sizes, int n_in,
                              void* d_out, int out_size, void* d_ws, size_t ws_size,
                              hipStream_t stream) {
    const float* x      = (const float*)d_in[0];
    const int*   idxA   = (const int*)d_in[1];
    const int*   idxC   = (const int*)d_in[2];
    const float* embA   = (const float*)d_in[3];
    const float* embC   = (const float*)d_in[4];
    const float* W1     = (const float*)d_in[5];
    const float* b1     = (const float*)d_in[6];
    const float* W2     = (const float*)d_in[7];
    const float* W3     = (const float*)d_in[8];
    const float* b3     = (const float*)d_in[9];
    float*       out    = (float*)d_out;

    float* Bs    = (float*)d_ws;                  // 4 * 2^18 floats = 4 MB
    float* Xs    = Bs + 4 * MAT_ELEMS;            // 2^19 floats     = 2 MB
    float* coef4 = Xs + BATCH * IN_DIM;           // 1024 * 4 floats

    coef_kernel<<<(BATCH + 255) / 256, 256, 0, stream>>>(
        idxA, idxC, embA, embC, W1, b1, W2, coef4);

    repack_B_kernel<<<(4 * MAT_ELEMS + 255) / 256, 256, 0, stream>>>(W3, b3, Bs);
    repack_A_kernel<<<(BATCH * IN_DIM + 255) / 256, 256, 0, stream>>>(x, Xs);

    // 1024 wave-tiles (32 Mblocks x 32 Ntiles), 4 waves/block -> 256 blocks
    gemm_basis<<<256, 128, 0, stream>>>(Xs, Bs, coef4, out);
}